// ClusterLossKLdiv_2284922601793
// MI455X (gfx1250) — compile-verified
//
#include <hip/hip_runtime.h>

typedef __attribute__((ext_vector_type(2))) float v2f;
typedef __attribute__((ext_vector_type(8))) float v8f;

#define N_ROWS   200000
#define D_IN_    256
#define D_LAT_   64
#define K_C      100
#define N_CT     7            // ceil(100/16) column tiles of 16
#define ALPHA_   1000.0

// ---------------------------------------------------------------------------
// Workspace layout: [0..7] double dec_accum, [8..15] double kl_accum,
//                   [16..463] float f[112]
// ---------------------------------------------------------------------------
__global__ void zero_ws_kernel(double* dec_acc, double* kl_acc, float* f) {
    if (threadIdx.x == 0) { *dec_acc = 0.0; *kl_acc = 0.0; }
    if (threadIdx.x < 112) f[threadIdx.x] = 0.0f;
}

// ---------------------------------------------------------------------------
// decoder_loss partial: sum (x-d)^2, float4 loads, f64 atomic per wave
// ---------------------------------------------------------------------------
__global__ void dec_loss_kernel(const float4* __restrict__ X,
                                const float4* __restrict__ D,
                                double* __restrict__ acc, int n4) {
    int i = blockIdx.x * blockDim.x + threadIdx.x;
    int stride = gridDim.x * blockDim.x;
    float s = 0.f;
    for (; i < n4; i += stride) {
        float4 x = X[i], d = D[i];
        float a = x.x - d.x, b = x.y - d.y, c = x.z - d.z, e = x.w - d.w;
        s += a * a + b * b + c * c + e * e;
    }
#pragma unroll
    for (int m = 1; m < 32; m <<= 1) s += __shfl_xor(s, m, 32);
    if ((threadIdx.x & 31) == 0) atomicAdd(acc, (double)s);
}

// ---------------------------------------------------------------------------
// Cluster kernel: one wave handles a 16-row strip of encoding.
//
// k-permutation trick: within one V_WMMA_F32_16X16X4_F32, lanes 0-15 feed
// WMMA-k {0,1} and lanes 16-31 feed WMMA-k {2,3}.  We map step t's
// WMMA-k {0,1} <- logical k {2t,2t+1} and WMMA-k {2,3} <- {32+2t,33+2t}.
// A and B use the same mapping, so the 64-term inner product is unchanged,
// and each lane's 16 fragments are 32 CONTIGUOUS floats -> 8 b128 loads.
//
// PHASE 1: accumulate f[k] = sum_n q[n,k].  PHASE 2: accumulate KL sum.
// 4 waves per 128-thread block, 3125 blocks -> exactly 12500 strips,
// every wave fully active => EXEC all ones at every WMMA.
// ---------------------------------------------------------------------------
template <int PHASE>
__global__ __launch_bounds__(128) void cluster_kernel(
    const float* __restrict__ enc, const float* __restrict__ cen,
    float* __restrict__ f, double* __restrict__ kl_acc) {

    const int wave = (blockIdx.x << 2) + (threadIdx.x >> 5);
    const int lane = threadIdx.x & 31;
    const int half = lane >> 4;        // 0: logical k 0..31 / rows 0..7 ; 1: k 32..63 / rows 8..15
    const int lm   = lane & 15;
    const int row0 = wave << 4;

    // ---- load A tile (16 x 64 of encoding) as 8 x b128 + row sq norms -----
    const float4* arow4 =
        (const float4*)(enc + (size_t)(row0 + lm) * D_LAT_ + half * 32);
    float4 av[8];
    float esum = 0.f;
#pragma unroll
    for (int j = 0; j < 8; ++j) {
        av[j] = arow4[j];
        esum += av[j].x * av[j].x + av[j].y * av[j].y +
                av[j].z * av[j].z + av[j].w * av[j].w;
    }
    esum += __shfl_xor(esum, 16, 32);          // full ||e_m||^2, m = lane&15

    v2f a[16];
#pragma unroll
    for (int j = 0; j < 8; ++j) {
        a[2 * j].x     = av[j].x;  a[2 * j].y     = av[j].y;
        a[2 * j + 1].x = av[j].z;  a[2 * j + 1].y = av[j].w;
    }

    // ---- GEMM: 7 column tiles of centres^T, 16 k-steps each ---------------
    v8f   acc[N_CT];
    float cn[N_CT];
#pragma unroll
    for (int ct = 0; ct < N_CT; ++ct) {
        const int  n  = ct * 16 + lm;
        const bool nv = (n < K_C);
        const float4* brow4 =
            (const float4*)(cen + (size_t)(nv ? n : 0) * D_LAT_ + half * 32);
        float4 bv[8];
        float csum = 0.f;
#pragma unroll
        for (int j = 0; j < 8; ++j) {
            bv[j] = brow4[j];
            if (!nv) { bv[j].x = 0.f; bv[j].y = 0.f; bv[j].z = 0.f; bv[j].w = 0.f; }
            csum += bv[j].x * bv[j].x + bv[j].y * bv[j].y +
                    bv[j].z * bv[j].z + bv[j].w * bv[j].w;
        }
        csum += __shfl_xor(csum, 16, 32);      // full ||c_n||^2

        v8f c = {};
#pragma unroll
        for (int j = 0; j < 8; ++j) {
            v2f b0, b1;
            b0.x = bv[j].x; b0.y = bv[j].y;
            b1.x = bv[j].z; b1.y = bv[j].w;
            c = __builtin_amdgcn_wmma_f32_16x16x4_f32(
                    false, a[2 * j],     false, b0, (short)0, c, false, false);
            c = __builtin_amdgcn_wmma_f32_16x16x4_f32(
                    false, a[2 * j + 1], false, b1, (short)0, c, false, false);
        }
        acc[ct] = c;
        cn[ct]  = csum;
    }

    // ---- broadcast row norms to the rows each lane owns in C --------------
    float en[8];
#pragma unroll
    for (int v = 0; v < 8; ++v) en[v] = __shfl(esum, v + half * 8, 32);

    // ---- q = 1/(1+d2), row-normalized -------------------------------------
    float rowsum[8];
#pragma unroll
    for (int v = 0; v < 8; ++v) rowsum[v] = 0.f;
#pragma unroll
    for (int ct = 0; ct < N_CT; ++ct) {
        const bool nv = (ct * 16 + lm) < K_C;
#pragma unroll
        for (int v = 0; v < 8; ++v) {
            float d2 = en[v] + cn[ct] - 2.f * acc[ct][v];
            d2 = fmaxf(d2, 0.f);
            float q = nv ? 1.f / (1.f + d2) : 0.f;
            acc[ct][v] = q;
            rowsum[v] += q;
        }
    }
#pragma unroll
    for (int v = 0; v < 8; ++v) {
#pragma unroll
        for (int m = 1; m < 16; m <<= 1)       // reduce within each half-wave
            rowsum[v] += __shfl_xor(rowsum[v], m, 32);
    }
#pragma unroll
    for (int ct = 0; ct < N_CT; ++ct)
#pragma unroll
        for (int v = 0; v < 8; ++v) acc[ct][v] = acc[ct][v] / rowsum[v];

    if (PHASE == 1) {
        // ---- f[k] += column sums of q --------------------------------------
#pragma unroll
        for (int ct = 0; ct < N_CT; ++ct) {
            float cs = 0.f;
#pragma unroll
            for (int v = 0; v < 8; ++v) cs += acc[ct][v];
            cs += __shfl_xor(cs, 16, 32);      // both 8-row halves of column n
            const int n = ct * 16 + lm;
            if (half == 0 && n < K_C) atomicAdd(&f[n], cs);
        }
    } else {
        // ---- p = (q^2/f) row-normalized; KL term sum -----------------------
        float fv[N_CT];
#pragma unroll
        for (int ct = 0; ct < N_CT; ++ct) {
            const int n = ct * 16 + lm;
            fv[ct] = f[(n < K_C) ? n : 0];
        }
        float prs[8];
#pragma unroll
        for (int v = 0; v < 8; ++v) prs[v] = 0.f;
#pragma unroll
        for (int ct = 0; ct < N_CT; ++ct)
#pragma unroll
            for (int v = 0; v < 8; ++v) {
                float q = acc[ct][v];
                prs[v] += q * q / fv[ct];      // q==0 for masked columns
            }
#pragma unroll
        for (int v = 0; v < 8; ++v) {
#pragma unroll
            for (int m = 1; m < 16; m <<= 1)
                prs[v] += __shfl_xor(prs[v], m, 32);
        }
        float term = 0.f;
#pragma unroll
        for (int ct = 0; ct < N_CT; ++ct)
#pragma unroll
            for (int v = 0; v < 8; ++v) {
                float q = acc[ct][v];
                if (q > 0.f) {
                    float p = (q * q / fv[ct]) / prs[v];
                    term += p * (logf(p) - logf(q));
                }
            }
#pragma unroll
        for (int m = 1; m < 32; m <<= 1) term += __shfl_xor(term, m, 32);
        if (lane == 0) atomicAdd(kl_acc, (double)term);
    }
}

// ---------------------------------------------------------------------------
__global__ void finalize_kernel(const double* dec_acc, const double* kl_acc,
                                float* out) {
    if (threadIdx.x == 0 && blockIdx.x == 0) {
        double dec = *dec_acc / ((double)N_ROWS * (double)D_IN_);
        double kl  = *kl_acc / (double)N_ROWS;
        out[0] = (float)(dec + ALPHA_ * kl);
    }
}

// ---------------------------------------------------------------------------
extern "C" void kernel_launch(void* const* d_in, const int* in_sizes, int n_in,
                              void* d_out, int out_size, void* d_ws, size_t ws_size,
                              hipStream_t stream) {
    const float* X    = (const float*)d_in[0];   // [200000,256]
    const float* enc  = (const float*)d_in[1];   // [200000,64]
    const float* decd = (const float*)d_in[2];   // [200000,256]
    const float* cen  = (const float*)d_in[3];   // [100,64]
    float* out = (float*)d_out;

    double* dec_acc = (double*)d_ws;
    double* kl_acc  = (double*)((char*)d_ws + 8);
    float*  f       = (float*)((char*)d_ws + 16);

    zero_ws_kernel<<<1, 128, 0, stream>>>(dec_acc, kl_acc, f);

    const int n4 = (N_ROWS * D_IN_) / 4;   // 12.8M float4 pairs
    dec_loss_kernel<<<2048, 256, 0, stream>>>(
        (const float4*)X, (const float4*)decd, dec_acc, n4);

    cluster_kernel<1><<<3125, 128, 0, stream>>>(enc, cen, f, kl_acc);
    cluster_kernel<2><<<3125, 128, 0, stream>>>(enc, cen, f, kl_acc);

    finalize_kernel<<<1, 32, 0, stream>>>(dec_acc, kl_acc, out);
}